// MultiResolutionHashEncoding_65025804862161
// MI455X (gfx1250) — compile-verified
//
#include <hip/hip_runtime.h>
#include <stdint.h>

// -----------------------------------------------------------------------------
// MultiResolutionHashEncoding for MI455X (gfx1250)
//
// Per block: loop over the 16 hash-table levels; stage each 128KB level table
// into LDS with the CDNA5 Tensor Data Mover (tensor_load_to_lds), double-
// buffered in 256KB of the 320KB WGP LDS so the DMA of level l+1 overlaps the
// LDS-gather + trilinear blend of level l. Gathers are ds_load_b64 against the
// 64-bank LDS instead of fully-divergent global loads (table >> 64KB WGP$).
// -----------------------------------------------------------------------------

#define NUM_LEVELS      16
#define TABLE_SZ        16384
#define FEAT            2
#define LEVEL_FLOATS    (TABLE_SZ * FEAT)          // 32768 floats per level
#define LEVEL_BYTES     (LEVEL_FLOATS * 4)         // 131072 bytes per level
#define THREADS         256
#define PTS_PER_THREAD  8
#define PTS_PER_BLOCK   (THREADS * PTS_PER_THREAD) // 2048

typedef __attribute__((ext_vector_type(4))) unsigned int tdm_g0_t; // D# group 0 (128b)
typedef __attribute__((ext_vector_type(8))) int          tdm_g1_t; // D# group 1 (256b)
typedef __attribute__((ext_vector_type(4))) int          tdm_g2_t; // D# groups 2/3 (128b)

// floor(16 * 1.38^l) computed in double, exact in f32
__device__ __constant__ float kRes[NUM_LEVELS] = {
    16.f,   22.f,   30.f,   42.f,   58.f,   80.f,   110.f,  152.f,
    210.f,  290.f,  400.f,  553.f,  763.f,  1053.f, 1453.f, 2005.f
};

// Issue one TDM transfer: 128KB level table (1-D tile of 32768 x 4B elements)
// from global memory into LDS at byte offset lds_byte_off.
__device__ __forceinline__ void tdm_load_level(const void* gsrc, unsigned lds_byte_off) {
    uint64_t ga = (uint64_t)(uintptr_t)gsrc;

    // ---- D# group 0 ----
    tdm_g0_t g0;
    g0.x = 1u;                                          // count=1 (valid descriptor)
    g0.y = lds_byte_off;                                // lds_addr [63:32]
    g0.z = (unsigned)ga;                                // global_addr[31:0]
    g0.w = (unsigned)((ga >> 32) & 0x01FFFFFFu)         // global_addr[56:32]
         | (2u << 30);                                  // type = 2 ("image")

    // ---- D# group 1 ----
    tdm_g1_t g1;
    g1[0] = 0x00020000;           // workgroup_mask=0, data_size=2 (4 bytes)
    g1[1] = (int)0x80000000u;     // tensor_dim0[15:0]=32768 in bits [31:16]
    g1[2] = 0x00010000;           // tensor_dim0[31:16]=0, tensor_dim1=1
    g1[3] = (int)0x80000000u;     // tensor_dim1 hi=0, tile_dim0=32768
    g1[4] = 0;                    // tile_dim1=0 (unused), tile_dim2=0
    g1[5] = LEVEL_FLOATS;         // tensor_dim0_stride low bits
    g1[6] = 0;
    g1[7] = 0;

    tdm_g2_t gz = {0, 0, 0, 0};   // groups 2/3 unused for a <=2-D descriptor

#if defined(__clang_major__) && (__clang_major__ >= 23)
    tdm_g1_t gz8 = {0, 0, 0, 0, 0, 0, 0, 0};
    __builtin_amdgcn_tensor_load_to_lds(g0, g1, gz, gz, gz8, 0);
#else
    __builtin_amdgcn_tensor_load_to_lds(g0, g1, gz, gz, 0);
#endif
}

__global__ __launch_bounds__(THREADS, 1)
void hashgrid_encode_kernel(const float* __restrict__ x,
                            const float* __restrict__ tables,
                            float* __restrict__ out,
                            int B) {
    extern __shared__ float2 lds_tab[];   // 2 x TABLE_SZ float2 = 256KB, double buffer

    // Flat shared-aperture address: low 32 bits are the LDS byte offset
    // (aperture maps addr[31:0] directly to LDS, same base as the D# lds_addr).
    const unsigned lds_base = (unsigned)(uintptr_t)(void*)lds_tab;

    const int tid  = (int)threadIdx.x;
    const int base = (int)blockIdx.x * PTS_PER_BLOCK;

    // ---- Preload this thread's points & fractional weights (reused all levels).
    // Loads are unguarded-but-clamped (x[B-1] is always valid) so the prologue
    // is a single branch-free clause of global_load_b96; only stores are guarded.
    float px[PTS_PER_THREAD], py[PTS_PER_THREAD], pz[PTS_PER_THREAD];
    float wx[PTS_PER_THREAD], wy[PTS_PER_THREAD], wz[PTS_PER_THREAD];
#pragma unroll
    for (int i = 0; i < PTS_PER_THREAD; ++i) {
        int b  = base + tid + i * THREADS;
        int bc = (b < B) ? b : (B - 1);
        float3 p = *(const float3*)(x + (size_t)bc * 3);
        px[i] = p.x; py[i] = p.y; pz[i] = p.z;
        // Faithful to reference: weights use the UNSCALED position
        wx[i] = p.x - floorf(p.x);
        wy[i] = p.y - floorf(p.y);
        wz[i] = p.z - floorf(p.z);
    }

    // ---- Prime the TDM pipeline: level 0 -> buffer 0 (one wave issues the DMA)
    if (tid < 32) tdm_load_level(tables, lds_base);

    for (int l = 0; l < NUM_LEVELS; ++l) {
        if (tid < 32) {
            if (l + 1 < NUM_LEVELS) {
                // Prefetch next level into the other buffer, then wait until
                // level l (the older outstanding TDM op) has landed.
                tdm_load_level(tables + (size_t)(l + 1) * LEVEL_FLOATS,
                               lds_base + (unsigned)((l + 1) & 1) * LEVEL_BYTES);
                __builtin_amdgcn_s_wait_tensorcnt((short)1);
            } else {
                __builtin_amdgcn_s_wait_tensorcnt((short)0);
            }
        }
        __syncthreads();   // level l data visible to the whole workgroup

        const float2* __restrict__ tab = lds_tab + (size_t)(l & 1) * TABLE_SZ;
        const float res = kRes[l];

#pragma unroll
        for (int i = 0; i < PTS_PER_THREAD; ++i) {
            int b = base + tid + i * THREADS;

            float sx = res * px[i], sy = res * py[i], sz = res * pz[i];
            // floor/ceil vertex coords as int32 (hi==lo when exactly integral);
            // hash factors precomputed once per axis (uint32 wraparound multiply
            // reproduces torch's int32 .int() arithmetic exactly).
            unsigned hxl = (unsigned)(int)floorf(sx);
            unsigned hxh = (unsigned)(int)ceilf(sx);
            unsigned hyl = 2654435761u * (unsigned)(int)floorf(sy);
            unsigned hyh = 2654435761u * (unsigned)(int)ceilf(sy);
            unsigned hzl = 805459861u  * (unsigned)(int)floorf(sz);
            unsigned hzh = 805459861u  * (unsigned)(int)ceilf(sz);

            float oxl = 1.f - wx[i], oyl = 1.f - wy[i], ozl = 1.f - wz[i];
            float a0 = 0.f, a1 = 0.f;

            // Corner order (x,y,z): 000,100,110,010,001,101,111,011
#pragma unroll
            for (int v = 0; v < 8; ++v) {
                const bool cx = (0x66u >> v) & 1u;
                const bool cy = (0xCCu >> v) & 1u;
                const bool cz = (0xF0u >> v) & 1u;
                unsigned h = (cx ? hxh : hxl) ^ (cy ? hyh : hyl) ^ (cz ? hzh : hzl);
                // TABLE_SZ is a power of two: two's-complement AND == python remainder
                float2 f = tab[h & (TABLE_SZ - 1)];         // ds_load_b64 gather
                float w = (cx ? wx[i] : oxl) * (cy ? wy[i] : oyl) * (cz ? wz[i] : ozl);
                a0 = fmaf(f.x, w, a0);
                a1 = fmaf(f.y, w, a1);
            }

            // out[b, l*2 + {0,1}] ; byte offset b*128 + l*8 (8B aligned)
            if (b < B) {
                *(float2*)(out + (size_t)b * (NUM_LEVELS * FEAT) + l * FEAT) =
                    make_float2(a0, a1);
            }
        }
        __syncthreads();   // all reads of buffer (l&1) done before it is re-filled
    }
}

extern "C" void kernel_launch(void* const* d_in, const int* in_sizes, int n_in,
                              void* d_out, int out_size, void* d_ws, size_t ws_size,
                              hipStream_t stream) {
    (void)n_in; (void)out_size; (void)d_ws; (void)ws_size;

    const float* x      = (const float*)d_in[0];   // [B,3] f32
    const float* tables = (const float*)d_in[1];   // [16,16384,2] f32
    float*       out    = (float*)d_out;           // [B,32] f32

    const int B = in_sizes[0] / 3;
    const int blocks = (B + PTS_PER_BLOCK - 1) / PTS_PER_BLOCK;

    hashgrid_encode_kernel<<<dim3(blocks), dim3(THREADS),
                             2 * LEVEL_BYTES, stream>>>(x, tables, out, B);
}